// Diagonal_26456998543635
// MI455X (gfx1250) — compile-verified
//
#include <hip/hip_runtime.h>

// out[i,j] = x[i,j] * |d[j]|   (x: 65536 x 1024 fp32, d: 1024 fp32)
//
// Pure streaming kernel: 512 MB HBM traffic, ~0.13 GFLOP -> bandwidth-bound
// by ~3 orders of magnitude (roofline floor ~22 us at 23.3 TB/s). No WMMA.
// Optimization targets: B128 vector memory ops with immediate offsets,
// TH=NT cache hints, zero per-iteration address arithmetic.

typedef __attribute__((ext_vector_type(4))) float f4;

// Static decomposition of the known problem size:
//   n4 = 65536*1024/4 = 16,777,216 float4 elements
//   4096 blocks x 256 threads x 16 float4/thread = 16,777,216 (exact cover)
#define TPB   256
#define ITERS 16
#define BLK_ELEMS (TPB * ITERS)   // 4096 float4 per block (contiguous 64 KB)

__global__ void __launch_bounds__(TPB)
diag_scale_kernel(const float* __restrict__ x,
                  const float* __restrict__ d,
                  float* __restrict__ out) {
    const unsigned t    = threadIdx.x;
    const unsigned base = blockIdx.x * BLK_ELEMS + t;   // float4 index

    // Row length = 1024 floats = 256 float4 columns. Intra-block stride is
    // 256 float4, so (base + k*256) & 255 == t for all k: the thread's
    // diagonal slice is iteration-invariant -> load |d| exactly once.
    const f4 dv = *(const f4*)(d + (size_t)(t & 255u) * 4);  // cached load (L2-resident 4 KB)
    f4 da;
    da.x = __builtin_fabsf(dv.x);
    da.y = __builtin_fabsf(dv.y);
    da.z = __builtin_fabsf(dv.z);
    da.w = __builtin_fabsf(dv.w);

    const f4* __restrict__ xv = (const f4*)x;
    f4* __restrict__ ov       = (f4*)out;

    // 16 iterations, byte offsets k*4096 from one shared base address:
    // the compiler folds these into the 24-bit IOFFSET immediate of
    // global_load_b128 / global_store_b128 -> no per-iteration VALU adds.
#pragma unroll
    for (int k = 0; k < ITERS; ++k) {
        const unsigned i = base + (unsigned)k * TPB;
        f4 v = __builtin_nontemporal_load(xv + i);
        v.x *= da.x;
        v.y *= da.y;
        v.z *= da.z;
        v.w *= da.w;
        __builtin_nontemporal_store(v, ov + i);
    }
}

extern "C" void kernel_launch(void* const* d_in, const int* in_sizes, int n_in,
                              void* d_out, int out_size, void* d_ws, size_t ws_size,
                              hipStream_t stream) {
    const float* x = (const float*)d_in[0];   // 65536*1024 fp32
    const float* d = (const float*)d_in[1];   // 1024 fp32

    const long long n4   = (long long)in_sizes[0] >> 2;        // 16,777,216
    const int       grid = (int)(n4 / BLK_ELEMS);              // 4096 blocks

    // 4096 blocks x 8 wave32 = 32K waves; each block streams a contiguous
    // 64 KB chunk of x -> out.
    diag_scale_kernel<<<grid, TPB, 0, stream>>>(x, d, (float*)d_out);
}